// ConvEnhanced_65481071404821
// MI455X (gfx1250) — compile-verified
//
#include <hip/hip_runtime.h>
#include <hip/hip_bf16.h>

// Fused conv2x2+relu + closed-form "quantum" map for MI455X (gfx1250).
//
// Key simplification (exact, not approximate): the 4-qubit circuit is
// RX-product-state -> (RZ diag + CNOT permutation) x2 -> <Z0>. Diagonal
// phases and permutations preserve Z-basis probabilities, so qparams drop
// out entirely and  expval = cos(a0)*cos(a1)*cos(a3)  where a0,a1,a3 are
// cls[2h,2w], cls[2h,2w+1], cls[2h+1,2w+1] (parity set {q0,q1,q3} comes
// from applying the 'full' CNOT bit-map twice: MSB(T^2(k)) = b0^b1^b3).
//
// The kernel is HBM-bound (~50 MB traffic). CDNA5-specific data path used:
// async global->LDS copies (ASYNCcnt) to stage each 33x33 x-tile once,
// giving 4x reuse and removing the stride-2 global read pattern.

#define QTILE      16          // 16x16 qvals per 256-thread block
#define XROWS      33          // x tile rows  (32 cls rows + 1 halo)
#define XCOLS      33          // x tile cols
#define LDS_STRIDE 48          // 2*LDS_STRIDE % 64 == 32 -> conflict-free stride-2 reads
#define XW         257         // x spatial size
#define OW         256         // cls spatial size

__global__ __launch_bounds__(256)
void fused_conv_quantum(const float* __restrict__ x,
                        const float* __restrict__ cw,
                        const float* __restrict__ cb,
                        float* __restrict__ out)
{
    __shared__ float tile[XROWS * LDS_STRIDE];

    const int b  = blockIdx.z;
    const int h0 = blockIdx.y * QTILE;      // qval-row base
    const int w0 = blockIdx.x * QTILE;      // qval-col base
    const int tid = threadIdx.x;

    const float* xb = x + (size_t)b * XW * XW;   // uniform 64-bit base -> SGPR pair

    // ---- stage x[2h0 .. 2h0+32][2w0 .. 2w0+32] into LDS via async copies ----
    // Low 32 bits of a generic pointer to __shared__ are the DS-relative address.
    const uint32_t lds_base = (uint32_t)(uintptr_t)(&tile[0]);
#pragma unroll
    for (int k = 0; k < 5; ++k) {
        int idx = tid + k * 256;
        if (idx < XROWS * XCOLS) {
            int r = idx / XCOLS;
            int c = idx - r * XCOLS;
            uint32_t lds_addr = lds_base + (uint32_t)((r * LDS_STRIDE + c) * 4);
            int      goff     = ((2 * h0 + r) * XW + (2 * w0 + c)) * 4;  // byte offset, GVS mode
            asm volatile("global_load_async_to_lds_b32 %0, %1, %2"
                         :: "v"(lds_addr), "v"(goff), "s"(xb)
                         : "memory");
        }
    }
    asm volatile("s_wait_asynccnt 0x0" ::: "memory");   // ASYNCcnt == 0: LDS writes landed
    __syncthreads();

    // ---- per-thread 2x2 conv + relu + closed-form expval ----
    const float w00 = cw[0], w01 = cw[1], w10 = cw[2], w11 = cw[3];
    const float bias = cb[0];

    const int tx = tid & (QTILE - 1);
    const int ty = tid >> 4;
    const float* t0 = &tile[(2 * ty) * LDS_STRIDE + 2 * tx];

    const float x00 = t0[0],              x01 = t0[1],              x02 = t0[2];
    const float x10 = t0[LDS_STRIDE],     x11 = t0[LDS_STRIDE + 1], x12 = t0[LDS_STRIDE + 2];
    const float x20 = t0[2 * LDS_STRIDE], x21 = t0[2 * LDS_STRIDE + 1], x22 = t0[2 * LDS_STRIDE + 2];

    float y00 = fmaxf(0.f, fmaf(w11, x11, fmaf(w10, x10, fmaf(w01, x01, fmaf(w00, x00, bias)))));
    float y01 = fmaxf(0.f, fmaf(w11, x12, fmaf(w10, x11, fmaf(w01, x02, fmaf(w00, x01, bias)))));
    float y10 = fmaxf(0.f, fmaf(w11, x21, fmaf(w10, x20, fmaf(w01, x11, fmaf(w00, x10, bias)))));
    float y11 = fmaxf(0.f, fmaf(w11, x22, fmaf(w10, x21, fmaf(w01, x12, fmaf(w00, x11, bias)))));

    // expval(Z0) of the 4-qubit circuit on angles (y00,y01,y10,y11): qparams cancel.
    const float q = cosf(y00) * cosf(y01) * cosf(y11);

    // ---- coalesced float2 stores: channel 0 = cls, channel 1 = qmap ----
    const int i = 2 * (h0 + ty);
    const int j = 2 * (w0 + tx);
    float* oc0 = out + (size_t)b * 2 * OW * OW;
    float* oc1 = oc0 + OW * OW;

    *(float2*)(oc0 + (size_t)i * OW + j)       = make_float2(y00, y01);
    *(float2*)(oc0 + (size_t)(i + 1) * OW + j) = make_float2(y10, y11);
    const float2 qq = make_float2(q, q);
    *(float2*)(oc1 + (size_t)i * OW + j)       = qq;
    *(float2*)(oc1 + (size_t)(i + 1) * OW + j) = qq;
}

extern "C" void kernel_launch(void* const* d_in, const int* in_sizes, int n_in,
                              void* d_out, int out_size, void* d_ws, size_t ws_size,
                              hipStream_t stream) {
    const float* x  = (const float*)d_in[0];   // [64,1,257,257]
    const float* cw = (const float*)d_in[1];   // [1,1,2,2]
    const float* cb = (const float*)d_in[2];   // [1]
    // d_in[3] (qparams) is provably unused: RZ phases + CNOT permutations
    // leave Z-basis probabilities permuted, and the permutation is constant.
    float* out = (float*)d_out;                // [64,2,256,256]

    dim3 grid(128 / QTILE, 128 / QTILE, 64);   // 8 x 8 x 64 = 4096 blocks
    fused_conv_quantum<<<grid, 256, 0, stream>>>(x, cw, cb, out);
}